// MaskedMultiHeadAttention_28312424415525
// MI455X (gfx1250) — compile-verified
//
#include <hip/hip_runtime.h>
#include <math.h>
#include <stdint.h>

typedef __attribute__((ext_vector_type(2))) float v2f;
typedef __attribute__((ext_vector_type(8))) float v8f;

#define S_LEN   1024
#define P_LEN   1024
#define B_SZ    4
#define E_DIM   1024
#define H_NUM   16
#define I_DIM   64
#define J_LEN   2048
#define HI_DIM  1024
#define SCHUNK  128
#define ATT_SCALE 0.125f   // 1/sqrt(I)

__device__ __forceinline__ v8f wmma4(v2f a, v2f b, v8f c) {
  // D = A(16x4) * B(4x16) + C(16x16), f32
  return __builtin_amdgcn_wmma_f32_16x16x4_f32(false, a, false, b, (short)0, c,
                                               false, false);
}

// ---------------------------------------------------------------------------
// Async global->LDS copy path (CDNA5 GLOBAL_LOAD_ASYNC_TO_LDS_B128, ASYNCcnt).
// Builtin signature (from clang diagnostic): (v4i AS1*, v4i AS3*, imm, imm)
// where v4i = __attribute__((vector_size(16))) int.
// ---------------------------------------------------------------------------
#if defined(__AMDGCN__) && __has_builtin(__builtin_amdgcn_global_load_async_to_lds_b128)
#define HAVE_ASYNC_LDS 1
#else
#define HAVE_ASYNC_LDS 0
#endif

typedef int v4i_vs __attribute__((vector_size(16)));
typedef __attribute__((address_space(1))) v4i_vs g1v4i;
typedef __attribute__((address_space(3))) v4i_vs l3v4i;

__device__ __forceinline__ void copy16_g2l(const float* g, float* l) {
#if HAVE_ASYNC_LDS
  // Generic LDS pointers carry the LDS byte offset in addr[31:0] (flat
  // aperture rule), so truncating to a 32-bit AS3 pointer is valid.
  __builtin_amdgcn_global_load_async_to_lds_b128(
      (g1v4i*)(uintptr_t)g, (l3v4i*)(uintptr_t)l, 0, 0);
#else
  *(float4*)l = *(const float4*)g;
#endif
}

template <int N>
__device__ __forceinline__ void async_wait() {
#if HAVE_ASYNC_LDS
#if __has_builtin(__builtin_amdgcn_s_wait_asynccnt)
  __builtin_amdgcn_s_wait_asynccnt(N);
#else
  asm volatile("s_wait_asynccnt %0" ::"i"(N) : "memory");
#endif
#endif
}

// ---------------------------------------------------------------------------
// Generic f32 GEMM: C[M,N] = A[M,K] @ B[K,N], row-major, dims % 64 == 0.
// 256 threads = 8 waves; block tile 64x64; wave tile 16x32; BK = 16.
// Double-buffered LDS fed by async global->LDS copies.
// A stride 20 dwords: 16B-aligned rows for B128 landing, 20m mod 64 spreads
// the 16 A-fragment row reads over distinct banks. B stride 64: reads are a
// row broadcast across consecutive columns (conflict-free).
// ---------------------------------------------------------------------------
__global__ __launch_bounds__(256) void gemm_f32(const float* __restrict__ A,
                                                const float* __restrict__ B,
                                                float* __restrict__ C,
                                                int M, int N, int K) {
  __shared__ float As[2][64 * 20];
  __shared__ float Bs[2][16 * 64];
  int tid  = threadIdx.x;
  int wave = tid >> 5, lane = tid & 31;
  int lo = lane & 15, hi = lane >> 4;
  int bm = blockIdx.x * 64, bn = blockIdx.y * 64;
  int wm = (wave & 3) * 16;
  int wn = (wave >> 2) * 32;

  // per-thread copy slots: A 64x16 tile (1x16B/lane), B 16x64 tile (1x16B/lane)
  int ra = tid >> 2, ca = (tid & 3) * 4;
  int rb = tid >> 4, cb = (tid & 15) * 4;
  const float* gA = A + (size_t)(bm + ra) * K + ca;
  const float* gB = B + (size_t)rb * N + bn + cb;
  float* lA[2] = {&As[0][ra * 20 + ca], &As[1][ra * 20 + ca]};
  float* lB[2] = {&Bs[0][rb * 64 + cb], &Bs[1][rb * 64 + cb]};
  size_t strideB = (size_t)16 * N;

  // prologue: stage 0 in flight
  copy16_g2l(gA, lA[0]);
  copy16_g2l(gB, lB[0]);

  v8f acc0 = {};
  v8f acc1 = {};
  int nstage = K >> 4;
  for (int st = 0; st < nstage; ++st) {
    int cur = st & 1;
    if (st + 1 < nstage) {   // issue next stage, then drain current
      copy16_g2l(gA + (size_t)(st + 1) * 16, lA[cur ^ 1]);
      copy16_g2l(gB + (size_t)(st + 1) * strideB, lB[cur ^ 1]);
      async_wait<2>();
    } else {
      async_wait<0>();
    }
    __syncthreads();
    const float* Ac = As[cur];
    const float* Bc = Bs[cur];
#pragma unroll
    for (int kk = 0; kk < 16; kk += 4) {
      int kc = kk + 2 * hi;
      float2 a2 = *(const float2*)(Ac + (wm + lo) * 20 + kc);
      v2f a; a[0] = a2.x; a[1] = a2.y;
      v2f b0; b0[0] = Bc[kc * 64 + wn + lo];      b0[1] = Bc[(kc + 1) * 64 + wn + lo];
      v2f b1; b1[0] = Bc[kc * 64 + wn + 16 + lo]; b1[1] = Bc[(kc + 1) * 64 + wn + 16 + lo];
      acc0 = wmma4(a, b0, acc0);
      acc1 = wmma4(a, b1, acc1);
    }
    __syncthreads();
  }
#pragma unroll
  for (int r = 0; r < 8; ++r) {
    int row = bm + wm + r + 8 * hi;
    C[(size_t)row * N + bn + wn + lo]      = acc0[r];
    C[(size_t)row * N + bn + wn + 16 + lo] = acc1[r];
  }
}

// ---------------------------------------------------------------------------
// Pos scores for one S-chunk, scattered pre-shifted:
//   posC[bh][s_loc][j] = (q[s,b,h,:]+v[h,:]) . p[j + (S-1-s), h, :]
// Grid: (B*H, SCHUNK/16); block 128 (4 waves, each a 16-wide j' strip).
// B fragments read straight from global p (J x 64 head slice stays L2-hot).
// ---------------------------------------------------------------------------
__global__ __launch_bounds__(128) void pos_chunk_kernel(
    const float* __restrict__ q, const float* __restrict__ p,
    const float* __restrict__ vmat, float* __restrict__ posC, int c0) {
  int bh = blockIdx.x;
  int b = bh >> 4, h = bh & 15;
  int tid = threadIdx.x;
  int wave = tid >> 5, lane = tid & 31;
  int lo = lane & 15, hi = lane >> 4;
  int s0loc = blockIdx.y * 16;

  v2f a_qv[16];
  {
    size_t qrow = ((size_t)(c0 + s0loc + lo) * B_SZ + b) * (size_t)HI_DIM +
                  (size_t)h * I_DIM;
#pragma unroll
    for (int t = 0; t < 16; ++t) {
      int col = 4 * t + 2 * hi;
      float2 qq = *(const float2*)(q + qrow + col);
      float2 vv = *(const float2*)(vmat + (size_t)h * I_DIM + col);
      a_qv[t][0] = qq.x + vv.x;
      a_qv[t][1] = qq.y + vv.y;
    }
  }
  for (int it = 0; it < J_LEN / 64; ++it) {
    int j0 = it * 64 + wave * 16;
    v8f acc = {};
#pragma unroll
    for (int t = 0; t < 16; ++t) {
      int kcol = 4 * t + 2 * hi;
      float2 pp = *(const float2*)(p + (size_t)(j0 + lo) * HI_DIM +
                                   (size_t)h * I_DIM + kcol);
      v2f bb; bb[0] = pp.x; bb[1] = pp.y;
      acc = wmma4(a_qv[t], bb, acc);
    }
#pragma unroll
    for (int r = 0; r < 8; ++r) {
      int s_local = s0loc + r + 8 * hi;
      int s_glob  = c0 + s_local;
      int j = (j0 + lo) - (S_LEN - 1) + s_glob;   // rel-shift scatter
      if (j >= 0)
        posC[((size_t)bh * SCHUNK + s_local) * J_LEN + j] = acc[r];
    }
  }
}

// ---------------------------------------------------------------------------
// Flash attention for one S-chunk. Grid: (B*H, SCHUNK/64); block 128.
// Each wave owns 16 query rows; K/V 16x64 tiles double-buffered in LDS and
// fed by async global->LDS copies (4 B128 copies per thread per stage).
// K/V stride 68 dwords: 272B rows (16B aligned for async landing) and
// 68 = 4 (mod 64) spreads the 16-lane K-fragment reads across banks.
// ---------------------------------------------------------------------------
__global__ __launch_bounds__(128) void attn_chunk_kernel(
    const float* __restrict__ q, const float* __restrict__ kv,
    const float* __restrict__ posC, const float* __restrict__ umat,
    float* __restrict__ awv, int c0) {
  __shared__ float Ks[2][16 * 68];
  __shared__ float Vs[2][16 * 68];
  __shared__ float Ps[4][16][18];
  int bh = blockIdx.x;
  int b = bh >> 4, h = bh & 15;
  int tid = threadIdx.x;
  int wave = tid >> 5, lane = tid & 31;
  int lo = lane & 15, hi = lane >> 4;
  int sblk_loc = blockIdx.y * 64;
  int sw_loc   = sblk_loc + wave * 16;

  v2f a_u[16];
  {
    size_t qrow = ((size_t)(c0 + sw_loc + lo) * B_SZ + b) * (size_t)HI_DIM +
                  (size_t)h * I_DIM;
#pragma unroll
    for (int t = 0; t < 16; ++t) {
      int col = 4 * t + 2 * hi;
      float2 qq = *(const float2*)(q + qrow + col);
      float2 uu = *(const float2*)(umat + (size_t)h * I_DIM + col);
      a_u[t][0] = qq.x + uu.x;
      a_u[t][1] = qq.y + uu.y;
    }
  }
  v8f acc_o[4] = {};
  float mrow[8], lrow[8];
#pragma unroll
  for (int r = 0; r < 8; ++r) { mrow[r] = -__builtin_inff(); lrow[r] = 0.f; }

  // per-thread K/V copy slots: 16 rows x 64 cols, 8 floats (2x16B) per thread
  int rk = tid >> 3, ck = (tid & 7) * 8;
  const float* gKV = kv + ((size_t)rk * B_SZ + b) * (size_t)(2 * HI_DIM) +
                     (size_t)h * I_DIM + ck;
  size_t stageStride = (size_t)16 * B_SZ * 2 * HI_DIM;   // 16 j-rows
  float* lK[2] = {&Ks[0][rk * 68 + ck], &Ks[1][rk * 68 + ck]};
  float* lV[2] = {&Vs[0][rk * 68 + ck], &Vs[1][rk * 68 + ck]};

  int ntiles = (P_LEN + c0 + sblk_loc + 64) >> 4;
  // prologue: stage 0 in flight
  {
    const float* g = gKV;
    copy16_g2l(g,              lK[0]);
    copy16_g2l(g + 4,          lK[0] + 4);
    copy16_g2l(g + HI_DIM,     lV[0]);
    copy16_g2l(g + HI_DIM + 4, lV[0] + 4);
  }

  for (int jt = 0; jt < ntiles; ++jt) {
    int cur = jt & 1;
    int j0 = jt * 16;
    if (jt + 1 < ntiles) {   // issue next K/V stage, drain current
      const float* g = gKV + (size_t)(jt + 1) * stageStride;
      copy16_g2l(g,              lK[cur ^ 1]);
      copy16_g2l(g + 4,          lK[cur ^ 1] + 4);
      copy16_g2l(g + HI_DIM,     lV[cur ^ 1]);
      copy16_g2l(g + HI_DIM + 4, lV[cur ^ 1] + 4);
      async_wait<4>();
    } else {
      async_wait<0>();
    }
    __syncthreads();
    const float* Kc = Ks[cur];
    const float* Vc = Vs[cur];

    v8f sc = {};
#pragma unroll
    for (int t = 0; t < 16; ++t) {
      int kcol = 4 * t + 2 * hi;
      float2 kk2 = *(const float2*)(Kc + lo * 68 + kcol);
      v2f bb; bb[0] = kk2.x; bb[1] = kk2.y;
      sc = wmma4(a_u[t], bb, sc);
    }
#pragma unroll
    for (int r = 0; r < 8; ++r) {
      int s_local = sw_loc + r + 8 * hi;
      int s_glob  = c0 + s_local;
      float pv = posC[((size_t)bh * SCHUNK + s_local) * J_LEN + j0 + lo];
      float t0 = (sc[r] + pv) * ATT_SCALE;
      bool valid = (j0 + lo) <= (P_LEN + s_glob);
      t0 = valid ? t0 : -__builtin_inff();
      float tmax = t0;
      tmax = fmaxf(tmax, __shfl_xor(tmax, 1, 32));
      tmax = fmaxf(tmax, __shfl_xor(tmax, 2, 32));
      tmax = fmaxf(tmax, __shfl_xor(tmax, 4, 32));
      tmax = fmaxf(tmax, __shfl_xor(tmax, 8, 32));
      float mnew  = fmaxf(mrow[r], tmax);
      float scale = __expf(mrow[r] - mnew);
      mrow[r] = mnew;
      float pe = __expf(t0 - mnew);
      float psum = pe;
      psum += __shfl_xor(psum, 1, 32);
      psum += __shfl_xor(psum, 2, 32);
      psum += __shfl_xor(psum, 4, 32);
      psum += __shfl_xor(psum, 8, 32);
      lrow[r] = lrow[r] * scale + psum;
#pragma unroll
      for (int f = 0; f < 4; ++f) acc_o[f][r] *= scale;
      Ps[wave][r + 8 * hi][lo] = pe;
    }
    // out += P(16x16) @ V(16x64)
#pragma unroll
    for (int t = 0; t < 4; ++t) {
      int kcol = 4 * t + 2 * hi;
      float2 pp2 = *(const float2*)&Ps[wave][lo][kcol];
      v2f ap; ap[0] = pp2.x; ap[1] = pp2.y;
#pragma unroll
      for (int f = 0; f < 4; ++f) {
        v2f bv;
        bv[0] = Vc[kcol * 68 + f * 16 + lo];
        bv[1] = Vc[(kcol + 1) * 68 + f * 16 + lo];
        acc_o[f] = wmma4(ap, bv, acc_o[f]);
      }
    }
    __syncthreads();
  }
#pragma unroll
  for (int r = 0; r < 8; ++r) {
    int s_glob = c0 + sw_loc + r + 8 * hi;
    float inv_l = 1.0f / lrow[r];
    size_t orow = ((size_t)s_glob * B_SZ + b) * (size_t)HI_DIM + (size_t)h * I_DIM;
#pragma unroll
    for (int f = 0; f < 4; ++f)
      awv[orow + f * 16 + lo] = acc_o[f][r] * inv_l;
  }
}

// ---------------------------------------------------------------------------
// out = LayerNorm(input + o) * gamma + beta, row length 1024. Grid S*B rows.
// ---------------------------------------------------------------------------
__global__ __launch_bounds__(256) void ln_kernel(const float* __restrict__ inp,
                                                 const float* __restrict__ o,
                                                 const float* __restrict__ gamma,
                                                 const float* __restrict__ beta,
                                                 float* __restrict__ out) {
  __shared__ float red[2][8];
  int row = blockIdx.x;
  float vals[4];
  float sum = 0.f, sumsq = 0.f;
#pragma unroll
  for (int it = 0; it < 4; ++it) {
    int e = it * 256 + threadIdx.x;
    float x = inp[(size_t)row * E_DIM + e] + o[(size_t)row * E_DIM + e];
    vals[it] = x; sum += x; sumsq += x * x;
  }
  for (int off = 1; off < 32; off <<= 1) {
    sum   += __shfl_xor(sum, off, 32);
    sumsq += __shfl_xor(sumsq, off, 32);
  }
  int wave = threadIdx.x >> 5, lane = threadIdx.x & 31;
  if (lane == 0) { red[0][wave] = sum; red[1][wave] = sumsq; }
  __syncthreads();
  float ts = 0.f, tq = 0.f;
#pragma unroll
  for (int i = 0; i < 8; ++i) { ts += red[0][i]; tq += red[1][i]; }
  float mean = ts * (1.0f / E_DIM);
  float var  = tq * (1.0f / E_DIM) - mean * mean;
  float rstd = rsqrtf(var + 1e-5f);
#pragma unroll
  for (int it = 0; it < 4; ++it) {
    int e = it * 256 + threadIdx.x;
    out[(size_t)row * E_DIM + e] = (vals[it] - mean) * rstd * gamma[e] + beta[e];
  }
}

// ---------------------------------------------------------------------------
extern "C" void kernel_launch(void* const* d_in, const int* in_sizes, int n_in,
                              void* d_out, int out_size, void* d_ws, size_t ws_size,
                              hipStream_t stream) {
  const float* inputMHA = (const float*)d_in[0];
  const float* posEmb   = (const float*)d_in[1];
  const float* memory   = (const float*)d_in[2];
  const float* umat     = (const float*)d_in[3];
  const float* vmat     = (const float*)d_in[4];
  // d_in[5] = mask (bool) -- causal mask recomputed analytically in-kernel
  const float* W_kv     = (const float*)d_in[6];
  const float* W_q      = (const float*)d_in[7];
  const float* W_p      = (const float*)d_in[8];
  const float* W_o      = (const float*)d_in[9];
  const float* gamma    = (const float*)d_in[10];
  const float* beta     = (const float*)d_in[11];
  float* out = (float*)d_out;

  float* ws    = (float*)d_ws;
  float* kvbuf = ws;                                     // 8192*2048
  float* qbuf  = kvbuf + (size_t)8192 * 2048;            // 4096*1024
  float* pbuf  = qbuf  + (size_t)4096 * 1024;            // 2048*1024
  float* posC  = pbuf  + (size_t)2048 * 1024;            // 64*SCHUNK*J
  float* awv   = posC  + (size_t)B_SZ * H_NUM * SCHUNK * J_LEN; // 4096*1024
  float* obuf  = awv   + (size_t)4096 * 1024;            // 4096*1024

  // Projections: kv (memory rows then input rows), q, p
  gemm_f32<<<dim3(64, 32), 256, 0, stream>>>(memory,   W_kv, kvbuf, 4096, 2048, 1024);
  gemm_f32<<<dim3(64, 32), 256, 0, stream>>>(inputMHA, W_kv, kvbuf + (size_t)4096 * 2048,
                                             4096, 2048, 1024);
  gemm_f32<<<dim3(64, 16), 256, 0, stream>>>(inputMHA, W_q, qbuf, 4096, 1024, 1024);
  gemm_f32<<<dim3(32, 16), 256, 0, stream>>>(posEmb,   W_p, pbuf, 2048, 1024, 1024);

  // Chunked pos-score + flash attention (stream order = dependency)
  for (int c = 0; c < S_LEN / SCHUNK; ++c) {
    pos_chunk_kernel<<<dim3(B_SZ * H_NUM, SCHUNK / 16), 128, 0, stream>>>(
        qbuf, pbuf, vmat, posC, c * SCHUNK);
    attn_chunk_kernel<<<dim3(B_SZ * H_NUM, SCHUNK / 64), 128, 0, stream>>>(
        qbuf, kvbuf, posC, umat, awv, c * SCHUNK);
  }

  // Output projection + residual + LayerNorm
  gemm_f32<<<dim3(64, 16), 256, 0, stream>>>(awv, W_o, obuf, 4096, 1024, 1024);
  ln_kernel<<<dim3(4096), 256, 0, stream>>>(inputMHA, obuf, gamma, beta, out);
}